// MergeRefineOracle4Param_52390011077285
// MI455X (gfx1250) — compile-verified
//
#include <hip/hip_runtime.h>
#include <hip/hip_bf16.h>
#include <math.h>

// Problem constants (from reference)
#define BB 8
#define CC 3
#define HH 512
#define WW 512
#define KK 8
#define MAX_SHIFT 0.5f

#define CHW   (CC * HH * WW)              // 786432
#define BCHW  (BB * CHW)                  // 6291456
#define AWSZ  (BB * KK * CHW)             // 50331648

// d_out float offsets (tuple concatenated in return order)
#define OFF_BW    0                        // best_warped  [B,C,H,W]
#define OFF_BIDX  (OFF_BW + BCHW)          // best_idx     [B]
#define OFF_BMSE  (OFF_BIDX + BB)          // best_mse     [B]
#define OFF_AW    (OFF_BMSE + BB)          // all_warped_b [B,K,C,H,W]
#define OFF_AMSE  (OFF_AW + AWSZ)          // all_mse      [B,K]
#define OFF_FLOW  (OFF_AMSE + BB * KK)     // flows        [K,H,W,2]

typedef __attribute__((ext_vector_type(2))) float v2f;
typedef __attribute__((ext_vector_type(8))) float v8f;

__device__ __forceinline__ float softplusf(float x) {
    return (x > 20.0f) ? x : log1pf(expf(x));
}

// ---------------------------------------------------------------------------
// K0: per-class separable flow tables. dxrow[k][y] = u_top*exp(-decay_y*y/(H-1))
//     dycol[k][x] = v_left*exp(-decay_x*x/(W-1))
// ---------------------------------------------------------------------------
__global__ void build_tables(const float* __restrict__ cp,
                             float* __restrict__ dxrow,
                             float* __restrict__ dycol) {
    const int k = blockIdx.x;
    const int t = threadIdx.x;
    const float vleft = MAX_SHIFT * tanhf(cp[k * 4 + 0]);
    const float utop  = MAX_SHIFT * tanhf(cp[k * 4 + 1]);
    const float decx  = softplusf(cp[k * 4 + 2]);
    const float decy  = softplusf(cp[k * 4 + 3]);
    if (t < HH) dxrow[k * HH + t] = utop  * expf(-decy * ((float)t * (1.0f / (HH - 1))));
    if (t < WW) dycol[k * WW + t] = vleft * expf(-decx * ((float)t * (1.0f / (WW - 1))));
}

// ---------------------------------------------------------------------------
// K1: materialize flows [K,H,W,2] from the separable tables (float2 stores).
// ---------------------------------------------------------------------------
__global__ void write_flows(const float* __restrict__ dxrow,
                            const float* __restrict__ dycol,
                            float* __restrict__ flows) {
    const int y = blockIdx.x, k = blockIdx.y, tid = threadIdx.x;
    const float dxk = dxrow[k * HH + y];
    float2* frow = (float2*)(flows + (((size_t)k * HH + y) * WW) * 2);
#pragma unroll
    for (int j = 0; j < 2; ++j) {
        const int x = tid + j * 256;
        frow[x] = make_float2(dxk, dycol[k * WW + x]);
    }
}

// ---------------------------------------------------------------------------
// K2: warp + squared-error partials. One workgroup per (k,y) handles the full
// row for all (b,c): shared weights, constant-per-row x-shift, |shift|<=0.5 so
// only rows y-1..y+1 are touched (L2-resident). all_warped via NT stores.
// ---------------------------------------------------------------------------
__global__ void warp_kernel(const float* __restrict__ pred,
                            const float* __restrict__ gt,
                            const float* __restrict__ dxrow,
                            const float* __restrict__ dycol,
                            float* __restrict__ all_warped,
                            float* __restrict__ partial) {
    const int y = blockIdx.x, k = blockIdx.y, tid = threadIdx.x;
    __shared__ float red[256];

    const float dxk = dxrow[k * HH + y];
    int xs[2], x0[2], x1[2], y0[2], y1[2];
    float wx[2], wy[2];
#pragma unroll
    for (int j = 0; j < 2; ++j) {
        const int x = tid + j * 256;
        xs[j] = x;
        float ix = fminf(fmaxf((float)x + dxk, 0.0f), (float)(WW - 1));
        float x0f = floorf(ix);
        x0[j] = (int)x0f;
        x1[j] = min(x0[j] + 1, WW - 1);
        wx[j] = ix - x0f;
        float iy = fminf(fmaxf((float)y + dycol[k * WW + x], 0.0f), (float)(HH - 1));
        float y0f = floorf(iy);
        y0[j] = (int)y0f;
        y1[j] = min(y0[j] + 1, HH - 1);
        wy[j] = iy - y0f;
    }

    for (int b = 0; b < BB; ++b) {
        if (b + 1 < BB) {
            // CDNA5 global_prefetch_b8: pull next batch's rows toward caches
            const float* nf = pred + (size_t)((b + 1) * CC) * HH * WW + (size_t)y * WW + xs[0];
            __builtin_prefetch(nf, 0, 0);
        }
        float err = 0.0f;
        for (int c = 0; c < CC; ++c) {
            const float* img  = pred + (size_t)(b * CC + c) * HH * WW;
            const float* grow = gt + ((size_t)(b * CC + c) * HH + y) * WW;
            float* orow = all_warped + ((((size_t)(b * KK + k)) * CC + c) * HH + y) * WW;
#pragma unroll
            for (int j = 0; j < 2; ++j) {
                const float* r0 = img + (size_t)y0[j] * WW;
                const float* r1 = img + (size_t)y1[j] * WW;
                const float v00 = r0[x0[j]], v01 = r0[x1[j]];
                const float v10 = r1[x0[j]], v11 = r1[x1[j]];
                const float top = v00 + wx[j] * (v01 - v00);
                const float bot = v10 + wx[j] * (v11 - v10);
                const float val = top + wy[j] * (bot - top);
                __builtin_nontemporal_store(val, &orow[xs[j]]);
                const float d = val - grow[xs[j]];
                err = fmaf(d, d, err);
            }
        }
        // deterministic block tree-reduce of err -> partial[(b*K+k)*H + y]
        red[tid] = err;
        __syncthreads();
        for (int s = 128; s > 0; s >>= 1) {
            if (tid < s) red[tid] += red[tid + s];
            __syncthreads();
        }
        if (tid == 0) partial[((size_t)(b * KK) + k) * HH + y] = red[0];
        __syncthreads();
    }
}

// ---------------------------------------------------------------------------
// K3: reduce 512 row-partials per (b,k) -> all_mse. One wave32 per (b,k);
// final cross-lane sum via V_WMMA_F32_16X16X4_F32 (B = 0.5 everywhere gives
// D[m][n] = s_m + s_{m+16}), 8 adds + one shfl_xor(16) complete the sum.
// ---------------------------------------------------------------------------
__global__ void mse_reduce(const float* __restrict__ partial,
                           float* __restrict__ all_mse) {
    const int bk = blockIdx.x;        // b*K + k
    const int lane = threadIdx.x;     // 0..31, wave32, EXEC all ones
    const float* p = partial + (size_t)bk * HH;
    float s = 0.0f;
    for (int i = lane; i < HH; i += 32) s += p[i];

#if __has_builtin(__builtin_amdgcn_wmma_f32_16x16x4_f32)
    v2f a;  a[0] = s;    a[1] = s;
    v2f bm; bm[0] = 0.5f; bm[1] = 0.5f;
    v8f acc = {};
    acc = __builtin_amdgcn_wmma_f32_16x16x4_f32(false, a, false, bm,
                                                (short)0, acc, false, false);
    float t = acc[0] + acc[1] + acc[2] + acc[3] + acc[4] + acc[5] + acc[6] + acc[7];
    t += __shfl_xor(t, 16, 32);
#else
    float t = s;
    for (int off = 16; off > 0; off >>= 1) t += __shfl_xor(t, off, 32);
#endif
    if (lane == 0) all_mse[bk] = t * (1.0f / (float)CHW);
}

// ---------------------------------------------------------------------------
// K4: per-batch argmin over K=8 (first-min tie-break like jnp.argmin).
// ---------------------------------------------------------------------------
__global__ void argmin_kernel(const float* __restrict__ all_mse,
                              float* __restrict__ bidx_f,
                              float* __restrict__ bmse,
                              int* __restrict__ bidx_i) {
    const int b = threadIdx.x;
    if (b < BB) {
        float best = all_mse[b * KK];
        int idx = 0;
        for (int k = 1; k < KK; ++k) {
            const float v = all_mse[b * KK + k];
            if (v < best) { best = v; idx = k; }
        }
        bidx_f[b] = (float)idx;
        bmse[b]   = best;
        bidx_i[b] = idx;
    }
}

// ---------------------------------------------------------------------------
// K5: gather best_warped[b] = all_warped[b, best_idx[b]] (float4 copies).
// ---------------------------------------------------------------------------
__global__ void gather_best(const float* __restrict__ all_warped,
                            const int* __restrict__ bidx,
                            float* __restrict__ best_warped) {
    const size_t CHW4 = (size_t)CHW / 4;
    const size_t i = (size_t)blockIdx.x * blockDim.x + threadIdx.x;
    if (i >= (size_t)BB * CHW4) return;
    const size_t b = i / CHW4, r = i % CHW4;
    const float4* src = (const float4*)all_warped + ((size_t)b * KK + bidx[b]) * CHW4 + r;
    ((float4*)best_warped)[i] = *src;
}

// ---------------------------------------------------------------------------
extern "C" void kernel_launch(void* const* d_in, const int* in_sizes, int n_in,
                              void* d_out, int out_size, void* d_ws, size_t ws_size,
                              hipStream_t stream) {
    const float* pred    = (const float*)d_in[0];
    const float* gt      = (const float*)d_in[1];
    const float* cparams = (const float*)d_in[2];

    float* out         = (float*)d_out;
    float* best_warped = out + OFF_BW;
    float* bidx_f      = out + OFF_BIDX;
    float* bmse        = out + OFF_BMSE;
    float* all_warped  = out + OFF_AW;
    float* all_mse     = out + OFF_AMSE;
    float* flows       = out + OFF_FLOW;

    float* ws      = (float*)d_ws;
    float* dxrow   = ws;                       // K*H
    float* dycol   = dxrow + KK * HH;          // K*W
    float* partial = dycol + KK * WW;          // B*K*H
    int*   bidx_i  = (int*)(partial + (size_t)BB * KK * HH);

    build_tables<<<KK, 512, 0, stream>>>(cparams, dxrow, dycol);
    write_flows<<<dim3(HH, KK), 256, 0, stream>>>(dxrow, dycol, flows);
    warp_kernel<<<dim3(HH, KK), 256, 0, stream>>>(pred, gt, dxrow, dycol,
                                                  all_warped, partial);
    mse_reduce<<<BB * KK, 32, 0, stream>>>(partial, all_mse);
    argmin_kernel<<<1, 32, 0, stream>>>(all_mse, bidx_f, bmse, bidx_i);
    const size_t nvec = (size_t)BB * (CHW / 4);
    gather_best<<<(unsigned)((nvec + 255) / 256), 256, 0, stream>>>(all_warped, bidx_i,
                                                                    best_warped);
}